// Denoise_Decoder3_45131516346834
// MI455X (gfx1250) — compile-verified
//
#include <hip/hip_runtime.h>
#include <math.h>

typedef float v2f __attribute__((ext_vector_type(2)));
typedef float v8f __attribute__((ext_vector_type(8)));

#define B_DIM    256
#define K_DIM    2048
#define D_DIM    1024
#define NTHREADS 256
#define NWAVES   8                         // 256 threads / wave32
#define ROWS_PER_WAVE (K_DIM / NWAVES)     // 256 k-rows per wave
#define CHAINS   4                         // 4 concurrent 16-row WMMA chains

__global__ __launch_bounds__(NTHREADS)
void attn_softmax_wmma(const float* __restrict__ q,
                       const float* __restrict__ cpt,
                       float* __restrict__ out) {
    __shared__ float s_q[D_DIM];          // 4 KB staged query
    __shared__ float s_p[K_DIM];          // 8 KB scores
    __shared__ float s_red[NTHREADS];     // 1 KB reduction scratch

    const int b    = blockIdx.x;
    const int t    = threadIdx.x;
    const int lane = t & 31;
    const int wave = t >> 5;

    const float* qb = q   + (size_t)b * D_DIM;
    const float* cb = cpt + (size_t)b * K_DIM * D_DIM;

    // Stage query into LDS (broadcast source for WMMA B fragments).
    for (int d = t; d < D_DIM; d += NTHREADS) s_q[d] = qb[d];
    __syncthreads();

    // ---------------- Pass A: scores via V_WMMA_F32_16X16X4_F32 ----------------
    // A fragment (16x4 f32): lanes 0-15 hold (M=lane, K=0..1), lanes 16-31 hold
    // (M=lane-16, K=2..3)  -> per-lane float2 load at row*D + d + (lane>>4)*2.
    // B fragment (4x16 f32), query replicated across N (value independent of
    // lane&15): one ds_load_b64 shared by all CHAINS wmma chains per d-step.
    const int rowInGrp = lane & 15;
    const int colOff   = (lane >> 4) * 2; // 0 or 2
    const float* qCol  = s_q + colOff;

    for (int gi = 0; gi < ROWS_PER_WAVE / (16 * CHAINS); ++gi) {  // 4 iters
        const int k0 = wave * ROWS_PER_WAVE + gi * (16 * CHAINS);

        v8f acc[CHAINS];
        const float* aRow[CHAINS];
#pragma unroll
        for (int c = 0; c < CHAINS; ++c) {
            acc[c]  = (v8f){0.f, 0.f, 0.f, 0.f, 0.f, 0.f, 0.f, 0.f};
            aRow[c] = cb + (size_t)(k0 + 16 * c + rowInGrp) * D_DIM + colOff;
        }

#pragma unroll 2
        for (int d = 0; d < D_DIM; d += 4) {
            const v2f bf = *(const v2f*)(qCol + d);   // ds_load_b64, shared
#pragma unroll
            for (int c = 0; c < CHAINS; ++c) {
                const v2f a = *(const v2f*)(aRow[c] + d);  // global_load_b64
                acc[c] = __builtin_amdgcn_wmma_f32_16x16x4_f32(
                             false, a, false, bf, (short)0, acc[c], false, false);
            }
        }

        // Column N=0 of each C tile holds its 16 dot products:
        // lane 0 -> M=0..7 in VGPR 0..7 ; lane 16 -> M=8..15 in VGPR 0..7.
        const float scale = 0.03125f; // 1/sqrt(1024)
        if (lane == 0) {
#pragma unroll
            for (int c = 0; c < CHAINS; ++c)
#pragma unroll
                for (int r = 0; r < 8; ++r)
                    s_p[k0 + 16 * c + r] = acc[c][r] * scale;
        } else if (lane == 16) {
#pragma unroll
            for (int c = 0; c < CHAINS; ++c)
#pragma unroll
                for (int r = 0; r < 8; ++r)
                    s_p[k0 + 16 * c + 8 + r] = acc[c][r] * scale;
        }
    }
    __syncthreads();

    // ---------------- Softmax over K in LDS ----------------
    float m = -INFINITY;
#pragma unroll
    for (int i = 0; i < K_DIM / NTHREADS; ++i)
        m = fmaxf(m, s_p[t + NTHREADS * i]);
    s_red[t] = m;
    __syncthreads();
    for (int s = NTHREADS / 2; s > 0; s >>= 1) {
        if (t < s) s_red[t] = fmaxf(s_red[t], s_red[t + s]);
        __syncthreads();
    }
    const float gmax = s_red[0];
    __syncthreads();

    float lsum = 0.f;
#pragma unroll
    for (int i = 0; i < K_DIM / NTHREADS; ++i) {
        const int idx = t + NTHREADS * i;
        const float p = __expf(s_p[idx] - gmax);
        s_p[idx] = p;
        lsum += p;
    }
    s_red[t] = lsum;
    __syncthreads();
    for (int s = NTHREADS / 2; s > 0; s >>= 1) {
        if (t < s) s_red[t] += s_red[t + s];
        __syncthreads();
    }
    const float inv = 1.0f / s_red[0];
    __syncthreads();

    // ---------------- Pass B: out[b,d] = inv * sum_k p[k]*C[b,k,d] ----------------
    // Each thread owns 4 contiguous columns -> fully coalesced float4 streams;
    // the 8 MB per-b slice re-read is largely served from the 192 MB L2.
    const float* colBase = cb + 4 * t;
    float4 acc = make_float4(0.f, 0.f, 0.f, 0.f);
    for (int k = 0; k < K_DIM; k += 4) {
        const float4 pv = *(const float4*)(s_p + k);
        const float4 c0 = *(const float4*)(colBase + (size_t)(k + 0) * D_DIM);
        const float4 c1 = *(const float4*)(colBase + (size_t)(k + 1) * D_DIM);
        const float4 c2 = *(const float4*)(colBase + (size_t)(k + 2) * D_DIM);
        const float4 c3 = *(const float4*)(colBase + (size_t)(k + 3) * D_DIM);
        acc.x += pv.x * c0.x + pv.y * c1.x + pv.z * c2.x + pv.w * c3.x;
        acc.y += pv.x * c0.y + pv.y * c1.y + pv.z * c2.y + pv.w * c3.y;
        acc.z += pv.x * c0.z + pv.y * c1.z + pv.z * c2.z + pv.w * c3.z;
        acc.w += pv.x * c0.w + pv.y * c1.w + pv.z * c2.w + pv.w * c3.w;
    }
    acc.x *= inv; acc.y *= inv; acc.z *= inv; acc.w *= inv;
    *(float4*)(out + (size_t)b * D_DIM + 4 * t) = acc;
}

extern "C" void kernel_launch(void* const* d_in, const int* in_sizes, int n_in,
                              void* d_out, int out_size, void* d_ws, size_t ws_size,
                              hipStream_t stream) {
    const float* q   = (const float*)d_in[0];   // query   [256, 1024]
    const float* cpt = (const float*)d_in[1];   // concept [256, 2048, 1024]
    float* out = (float*)d_out;                 // [256, 1024]
    dim3 grid(B_DIM), block(NTHREADS);
    attn_softmax_wmma<<<grid, block, 0, stream>>>(q, cpt, out);
}